// SGDrop_2345052143676
// MI455X (gfx1250) — compile-verified
//
#include <hip/hip_runtime.h>

// SGDrop for MI455X (gfx1250, wave32).
// attr = relu(grad*features); per-sample exact k-th largest; out = f*(attr<=thr).
//
// Fast path (2 streaming scans, minimal traffic ~1.03 GB):
//   1) 16384-bin histogram over bits[30:17] (64 KB LDS/block; 320 KB WGP pool)
//   2) fused apply+collect: out decided per-element by bucket compare; the few
//      in-bucket candidates (pos,bits,val) are recorded for later fixup
//   3) exact MSB-first select over remaining 17 bits (tiny)
//   4) scatter fixup of candidate positions only (~83K writes total)
// Fallback (ws too small): 3-level 4096/1024/1024 radix select + plain apply.

typedef float v4f __attribute__((ext_vector_type(4)));

#define SG_B        64
#define SG_N        802816              // 256*56*56 per sample
#define SG_NQ       (SG_N / 4)          // 200704 float4 per sample
#define SG_K        80281u              // int(0.1 * N)
#define SG_THREADS  256
#define SG_BLKS     112                 // 112*256 = 28672 threads/sample
#define SG_ITERS    7                   // 200704 / 28672 = 7 exactly
#define SG_STRIDE   (SG_BLKS * SG_THREADS)

// ---- shared workspace header (uint32 words) ----
#define WS_THRESH   0                   // [64] final threshold bits
#define WS_KREM     64                  // [64] rank within bucket
#define WS_PREFIX   128                 // [64] bucket prefix (bin << 17)
#define WS_CNT      192                 // [64] candidate counters
#define WS_HIST     256                 // histogram starts here

// ---- fast path ----
#define FBINS       16384               // bits[30:17]
#define FSHIFT      17
#define FCAP        8192                // candidate cap per sample (~6x headroom)
#define WS_CBITS    (WS_HIST + SG_B * FBINS)
#define WS_CPOS     (WS_CBITS + SG_B * FCAP)
#define WS_CVAL     (WS_CPOS + SG_B * FCAP)
#define WS_FAST_WORDS (WS_CVAL + SG_B * FCAP)

// ---- fallback path ----
#define SBINS0      4096

// ======================= fast path =======================

__global__ void sg_init_fast(unsigned* __restrict__ ws) {
    const int i = blockIdx.x * blockDim.x + threadIdx.x;
    const int total = SG_B * FBINS;
    for (int j = i; j < total; j += gridDim.x * blockDim.x) ws[WS_HIST + j] = 0u;
    if (i < SG_B) { ws[WS_CNT + i] = 0u; }
}

__global__ __launch_bounds__(SG_THREADS)
void sg_hist_fast(const float* __restrict__ f, const float* __restrict__ g,
                  unsigned* __restrict__ ws) {
    __shared__ unsigned lhist[FBINS];                       // 64 KB LDS
    const int sample = blockIdx.y;
    for (int i = threadIdx.x; i < FBINS; i += SG_THREADS) lhist[i] = 0u;
    __syncthreads();

    const v4f* f4 = (const v4f*)f + (size_t)sample * SG_NQ;
    const v4f* g4 = (const v4f*)g + (size_t)sample * SG_NQ;
    const int tid = blockIdx.x * SG_THREADS + threadIdx.x;

    unsigned zc = 0;
#pragma unroll
    for (int it = 0; it < SG_ITERS; ++it) {
        const int idx = it * SG_STRIDE + tid;
        if (it + 1 < SG_ITERS) {
            __builtin_prefetch(&f4[idx + SG_STRIDE], 0, 0);
            __builtin_prefetch(&g4[idx + SG_STRIDE], 0, 0);
        }
        v4f fv = __builtin_nontemporal_load(&f4[idx]);
        v4f gv = __builtin_nontemporal_load(&g4[idx]);
#pragma unroll
        for (int l = 0; l < 4; ++l) {
            const float p = fv[l] * gv[l];
            if (p > 0.0f) {
                const unsigned bits = __float_as_uint(p);   // bit31 == 0
                atomicAdd(&lhist[(bits >> FSHIFT) & (FBINS - 1)], 1u);
            } else {
                ++zc;                                        // relu -> 0.0, bin 0
            }
        }
    }
    if (zc) atomicAdd(&lhist[0], zc);
    __syncthreads();

    unsigned* gh = ws + WS_HIST + sample * FBINS;
    for (int i = threadIdx.x; i < FBINS; i += SG_THREADS) {
        const unsigned c = lhist[i];
        if (c) atomicAdd(&gh[i], c);
    }
}

__global__ __launch_bounds__(SG_THREADS)
void sg_select_fast(unsigned* __restrict__ ws) {
    __shared__ unsigned partial[SG_THREADS];
    const int sample = blockIdx.x;
    const unsigned* h = ws + WS_HIST + sample * FBINS;
    const int ch = FBINS / SG_THREADS;                      // 64 bins per thread

    unsigned s = 0;
    for (int j = 0; j < ch; ++j) s += h[threadIdx.x * ch + j];
    partial[threadIdx.x] = s;
    __syncthreads();

    if (threadIdx.x == 0) {
        unsigned cum = 0;
        int chunk = 0;
        for (int i = SG_THREADS - 1; i >= 0; --i) {
            if (cum + partial[i] >= SG_K) { chunk = i; break; }
            cum += partial[i];
        }
        int bin = chunk * ch;
        for (int j = ch - 1; j >= 0; --j) {
            const unsigned c = h[chunk * ch + j];
            if (cum + c >= SG_K) { bin = chunk * ch + j; break; }
            cum += c;
        }
        ws[WS_PREFIX + sample] = (unsigned)bin << FSHIFT;
        ws[WS_KREM + sample]   = SG_K - cum;                // rank within bucket
    }
}

// Fused apply + candidate collection. Elements outside the threshold bucket
// are fully decided here; in-bucket elements get a provisional "keep" and are
// recorded for the fixup pass.
__global__ __launch_bounds__(SG_THREADS)
void sg_apply_collect(const float* __restrict__ f, const float* __restrict__ g,
                      unsigned* __restrict__ ws, float* __restrict__ out) {
    const int sample = blockIdx.y;
    const unsigned pref = ws[WS_PREFIX + sample];           // bucket << 17
    const unsigned bhi  = pref + (1u << FSHIFT);            // bucket upper bound
    const v4f* f4 = (const v4f*)f + (size_t)sample * SG_NQ;
    const v4f* g4 = (const v4f*)g + (size_t)sample * SG_NQ;
    v4f* o4 = (v4f*)out + (size_t)sample * SG_NQ;
    unsigned* cbits = ws + WS_CBITS + (size_t)sample * FCAP;
    unsigned* cpos  = ws + WS_CPOS  + (size_t)sample * FCAP;
    unsigned* cval  = ws + WS_CVAL  + (size_t)sample * FCAP;
    unsigned* cnt   = ws + WS_CNT + sample;
    const int tid = blockIdx.x * SG_THREADS + threadIdx.x;
#pragma unroll
    for (int it = 0; it < SG_ITERS; ++it) {
        const int idx = it * SG_STRIDE + tid;
        if (it + 1 < SG_ITERS) {
            __builtin_prefetch(&f4[idx + SG_STRIDE], 0, 0);
            __builtin_prefetch(&g4[idx + SG_STRIDE], 0, 0);
        }
        v4f fv = __builtin_nontemporal_load(&f4[idx]);
        v4f gv = __builtin_nontemporal_load(&g4[idx]);
        v4f ov;
#pragma unroll
        for (int l = 0; l < 4; ++l) {
            const float p = fv[l] * gv[l];
            float o = fv[l];                                // default: keep
            if (p > 0.0f) {
                const unsigned bits = __float_as_uint(p);
                if (bits >= bhi) {
                    o = 0.0f;                               // above bucket: drop
                } else if (bits >= pref) {                  // in bucket: candidate
                    const unsigned slot = atomicAdd(cnt, 1u);
                    if (slot < FCAP) {
                        cbits[slot] = bits;
                        cpos[slot]  = (unsigned)(idx * 4 + l);
                        cval[slot]  = __float_as_uint(fv[l]);
                    }
                }
            }
            ov[l] = o;
        }
        __builtin_nontemporal_store(ov, &o4[idx]);
    }
}

// Exact MSB-first bit select over the low 17 bits among the candidates.
__global__ __launch_bounds__(SG_THREADS)
void sg_final(unsigned* __restrict__ ws) {
    __shared__ unsigned cl[FCAP];                           // 32 KB
    __shared__ unsigned red[SG_THREADS];
    const int sample = blockIdx.x;
    const unsigned* cand = ws + WS_CBITS + (size_t)sample * FCAP;
    unsigned cnt = ws[WS_CNT + sample];
    if (cnt > FCAP) cnt = FCAP;

    for (unsigned i = threadIdx.x; i < cnt; i += SG_THREADS)
        cl[i] = cand[i] & 0x1FFFFu;
    __syncthreads();

    unsigned prefix = 0;
    unsigned krem = ws[WS_KREM + sample];
    for (int b = 16; b >= 0; --b) {
        const unsigned bit = 1u << b;
        const unsigned hi_mask = 0x1FFFFu & ~((bit << 1) - 1u);
        unsigned c = 0;
        for (unsigned i = threadIdx.x; i < cnt; i += SG_THREADS) {
            const unsigned v = cl[i];
            if ((v & hi_mask) == prefix && (v & bit)) ++c;
        }
        red[threadIdx.x] = c;
        __syncthreads();
        for (int s = SG_THREADS / 2; s > 0; s >>= 1) {
            if (threadIdx.x < s) red[threadIdx.x] += red[threadIdx.x + s];
            __syncthreads();
        }
        const unsigned tot = red[0];
        __syncthreads();
        if (tot >= krem) prefix |= bit; else krem -= tot;
    }
    if (threadIdx.x == 0)
        ws[WS_THRESH + sample] = ws[WS_PREFIX + sample] | prefix;
}

// Rewrite only the candidate positions with the exact threshold decision.
__global__ __launch_bounds__(SG_THREADS)
void sg_fixup(const unsigned* __restrict__ ws, float* __restrict__ out) {
    const int sample = blockIdx.x;
    unsigned cnt = ws[WS_CNT + sample];
    if (cnt > FCAP) cnt = FCAP;
    const unsigned thr = ws[WS_THRESH + sample];
    const unsigned* cbits = ws + WS_CBITS + (size_t)sample * FCAP;
    const unsigned* cpos  = ws + WS_CPOS  + (size_t)sample * FCAP;
    const unsigned* cval  = ws + WS_CVAL  + (size_t)sample * FCAP;
    float* o = out + (size_t)sample * SG_N;
    for (unsigned i = threadIdx.x; i < cnt; i += SG_THREADS) {
        // uint compare == float compare for non-negative floats
        o[cpos[i]] = (cbits[i] <= thr) ? __uint_as_float(cval[i]) : 0.0f;
    }
}

// ======================= fallback path (3-level radix) =======================

__global__ void sg_init_slow(unsigned* __restrict__ ws) {
    const int i = blockIdx.x * blockDim.x + threadIdx.x;
    const int total = SG_B * SBINS0;
    for (int j = i; j < total; j += gridDim.x * blockDim.x) ws[WS_HIST + j] = 0u;
    if (i < SG_B) { ws[WS_PREFIX + i] = 0u; ws[WS_KREM + i] = SG_K; }
}

__global__ __launch_bounds__(SG_THREADS)
void sg_hist_slow(const float* __restrict__ f, const float* __restrict__ g,
                  unsigned* __restrict__ ws, int shift, int nbins, unsigned pmask) {
    __shared__ unsigned lhist[SBINS0];
    const int sample = blockIdx.y;
    for (int i = threadIdx.x; i < nbins; i += SG_THREADS) lhist[i] = 0u;
    __syncthreads();

    const unsigned pref = ws[WS_PREFIX + sample];
    const v4f* f4 = (const v4f*)f + (size_t)sample * SG_NQ;
    const v4f* g4 = (const v4f*)g + (size_t)sample * SG_NQ;
    const int tid = blockIdx.x * SG_THREADS + threadIdx.x;

    unsigned zc = 0;
#pragma unroll
    for (int it = 0; it < SG_ITERS; ++it) {
        const int idx = it * SG_STRIDE + tid;
        v4f fv = __builtin_nontemporal_load(&f4[idx]);
        v4f gv = __builtin_nontemporal_load(&g4[idx]);
#pragma unroll
        for (int l = 0; l < 4; ++l) {
            const float p = fv[l] * gv[l];
            if (p > 0.0f) {
                const unsigned bits = __float_as_uint(p);
                if ((bits & pmask) == pref)
                    atomicAdd(&lhist[(bits >> shift) & (unsigned)(nbins - 1)], 1u);
            } else {
                ++zc;
            }
        }
    }
    if (pref == 0u && zc) atomicAdd(&lhist[0], zc);
    __syncthreads();

    unsigned* gh = ws + WS_HIST + sample * SBINS0;
    for (int i = threadIdx.x; i < nbins; i += SG_THREADS) {
        const unsigned c = lhist[i];
        if (c) atomicAdd(&gh[i], c);
    }
}

__global__ __launch_bounds__(SG_THREADS)
void sg_select_slow(unsigned* __restrict__ ws, int nbins, int shift, int isFinal) {
    __shared__ unsigned partial[SG_THREADS];
    const int sample = blockIdx.x;
    unsigned* h = ws + WS_HIST + sample * SBINS0;
    const int ch = nbins / SG_THREADS;

    unsigned s = 0;
    for (int j = 0; j < ch; ++j) s += h[threadIdx.x * ch + j];
    partial[threadIdx.x] = s;
    __syncthreads();

    if (threadIdx.x == 0) {
        const unsigned krem = ws[WS_KREM + sample];
        unsigned cum = 0;
        int chunk = 0;
        for (int i = SG_THREADS - 1; i >= 0; --i) {
            if (cum + partial[i] >= krem) { chunk = i; break; }
            cum += partial[i];
        }
        int bin = chunk * ch;
        for (int j = ch - 1; j >= 0; --j) {
            const unsigned c = h[chunk * ch + j];
            if (cum + c >= krem) { bin = chunk * ch + j; break; }
            cum += c;
        }
        if (isFinal) {
            ws[WS_THRESH + sample] = ws[WS_PREFIX + sample] | ((unsigned)bin << shift);
        } else {
            ws[WS_PREFIX + sample] |= ((unsigned)bin) << shift;
            ws[WS_KREM  + sample]  = krem - cum;
        }
    }
    __syncthreads();
    for (int i = threadIdx.x; i < nbins; i += SG_THREADS) h[i] = 0u;
}

__global__ __launch_bounds__(SG_THREADS)
void sg_apply_plain(const float* __restrict__ f, const float* __restrict__ g,
                    const unsigned* __restrict__ ws, float* __restrict__ out) {
    const int sample = blockIdx.y;
    const float thr = __uint_as_float(ws[WS_THRESH + sample]);
    const v4f* f4 = (const v4f*)f + (size_t)sample * SG_NQ;
    const v4f* g4 = (const v4f*)g + (size_t)sample * SG_NQ;
    v4f* o4 = (v4f*)out + (size_t)sample * SG_NQ;
    const int tid = blockIdx.x * SG_THREADS + threadIdx.x;
#pragma unroll
    for (int it = 0; it < SG_ITERS; ++it) {
        const int idx = it * SG_STRIDE + tid;
        v4f fv = __builtin_nontemporal_load(&f4[idx]);
        v4f gv = __builtin_nontemporal_load(&g4[idx]);
        v4f ov;
#pragma unroll
        for (int l = 0; l < 4; ++l) {
            const float a = fmaxf(fv[l] * gv[l], 0.0f);
            ov[l] = (a <= thr) ? fv[l] : 0.0f;
        }
        __builtin_nontemporal_store(ov, &o4[idx]);
    }
}

// ======================= launcher =======================

extern "C" void kernel_launch(void* const* d_in, const int* in_sizes, int n_in,
                              void* d_out, int out_size, void* d_ws, size_t ws_size,
                              hipStream_t stream) {
    (void)in_sizes; (void)n_in; (void)out_size;
    const float* f = (const float*)d_in[0];
    const float* g = (const float*)d_in[1];
    float* out = (float*)d_out;
    unsigned* ws = (unsigned*)d_ws;

    const dim3 grid(SG_BLKS, SG_B);

    if (ws_size >= (size_t)WS_FAST_WORDS * sizeof(unsigned)) {
        sg_init_fast    <<<1024, SG_THREADS, 0, stream>>>(ws);
        sg_hist_fast    <<<grid, SG_THREADS, 0, stream>>>(f, g, ws);
        sg_select_fast  <<<SG_B, SG_THREADS, 0, stream>>>(ws);
        sg_apply_collect<<<grid, SG_THREADS, 0, stream>>>(f, g, ws, out);
        sg_final        <<<SG_B, SG_THREADS, 0, stream>>>(ws);
        sg_fixup        <<<SG_B, SG_THREADS, 0, stream>>>(ws, out);
    } else {
        sg_init_slow  <<<256, SG_THREADS, 0, stream>>>(ws);
        sg_hist_slow  <<<grid, SG_THREADS, 0, stream>>>(f, g, ws, 20, 4096, 0x00000000u);
        sg_select_slow<<<SG_B, SG_THREADS, 0, stream>>>(ws, 4096, 20, 0);
        sg_hist_slow  <<<grid, SG_THREADS, 0, stream>>>(f, g, ws, 10, 1024, 0xFFF00000u);
        sg_select_slow<<<SG_B, SG_THREADS, 0, stream>>>(ws, 1024, 10, 0);
        sg_hist_slow  <<<grid, SG_THREADS, 0, stream>>>(f, g, ws,  0, 1024, 0xFFFFFC00u);
        sg_select_slow<<<SG_B, SG_THREADS, 0, stream>>>(ws, 1024,  0, 1);
        sg_apply_plain<<<grid, SG_THREADS, 0, stream>>>(f, g, ws, out);
    }
}